// HeterophilyGNNv2_52115133169690
// MI455X (gfx1250) — compile-verified
//
#include <hip/hip_runtime.h>

typedef __attribute__((ext_vector_type(2))) float v2f;
typedef __attribute__((ext_vector_type(8))) float v8f;
typedef unsigned __attribute__((ext_vector_type(4))) u32x4;
typedef int __attribute__((ext_vector_type(4))) i32x4;
typedef int __attribute__((ext_vector_type(8))) i32x8;

#define NEG_SLOPE 0.2f
#define LN_EPS    1e-5f
#define SM_EPS    1e-16f

#ifndef __has_builtin
#define __has_builtin(x) 0
#endif
#if __has_builtin(__builtin_amdgcn_tensor_load_to_lds) && \
    __has_builtin(__builtin_amdgcn_s_wait_tensorcnt)
#define HAVE_TDM 1
#else
#define HAVE_TDM 0
#endif
#if HAVE_TDM
#if defined(__has_include)
#if __has_include(<hip/amd_detail/amd_gfx1250_TDM.h>)
#define TDM_SIX_ARGS 1
#endif
#endif
#ifndef TDM_SIX_ARGS
#define TDM_SIX_ARGS 0
#endif
#if TDM_SIX_ARGS
#define TDM_LOAD(g0, g1, g2, g3) \
  __builtin_amdgcn_tensor_load_to_lds(g0, g1, g2, g3, (i32x8)(0), 0)
#else
#define TDM_LOAD(g0, g1, g2, g3) \
  __builtin_amdgcn_tensor_load_to_lds(g0, g1, g2, g3, 0)
#endif
#endif

// ---- order-preserving float <-> uint encoding for atomicMax on floats ----
__device__ __forceinline__ unsigned enc_f(float f) {
  unsigned u = __float_as_uint(f);
  return (u & 0x80000000u) ? ~u : (u | 0x80000000u);
}
__device__ __forceinline__ float dec_f(unsigned u) {
  return __uint_as_float((u & 0x80000000u) ? (u & 0x7fffffffu) : ~u);
}
__device__ __forceinline__ float wave_sum(float v) {
#pragma unroll
  for (int o = 16; o > 0; o >>= 1) v += __shfl_xor(v, o, 32);
  return v;
}

// ===========================================================================
// Dual GEMM: Yl = X @ Wl, Yr = X @ Wr   (X:[nrows,FIN], W:[FIN,FOUT])
// Block = 16 rows, 8 waves. X tile staged into LDS by the Tensor Data Mover
// (one DMA per block; D# pad fields implement the +4-dword anti-conflict row
// padding; TDM OOB semantics zero-fill the ragged last block). Each wave owns
// 16x16 output tiles, walking K in steps of 4 with V_WMMA_F32_16X16X4_F32.
// ===========================================================================
template <int FIN, int FOUT>
__global__ void __launch_bounds__(256) gemm_pair(
    const float* __restrict__ X, const float* __restrict__ Wl,
    const float* __restrict__ Wr, float* __restrict__ Yl,
    float* __restrict__ Yr, int nrows) {
  constexpr int LDSS = FIN + 4;  // +4 dwords: bank = (4*m + k) % 64, conflict-free
  __shared__ float xs[16 * LDSS];
  const int row0 = blockIdx.x * 16;
  const int tid = threadIdx.x;

#if HAVE_TDM
  if (tid < 32) {  // wave 0 drives the TDM
    unsigned lds_base =
        (unsigned)(size_t)(__attribute__((address_space(3))) float*)&xs[0];
    unsigned long long ga =
        (unsigned long long)(size_t)(X + (size_t)row0 * FIN);
    unsigned rem = (unsigned)(nrows - row0);  // tensor rows left; OOB -> 0
    constexpr unsigned PI = (FIN == 256) ? 7u : 5u;  // pad every FIN dwords
    constexpr unsigned PA = 3u;                      // pad 4 dwords
    u32x4 g0 = {1u,                    // count=1, user mode, no gather
                lds_base,              // lds_addr
                (unsigned)ga,          // global_addr[31:0]
                (unsigned)(ga >> 32) | 0x80000000u};  // addr[56:32] | type=2
    i32x8 g1 = {
        (int)((2u << 16) | (1u << 20) | (PI << 22) | (PA << 25)),  // dsz=4B,pad
        (int)(((unsigned)FIN & 0xFFFFu) << 16),       // tensor_dim0 lo
        (int)((rem & 0xFFFFu) << 16),                 // dim0 hi(0) | dim1 lo
        (int)(((rem >> 16) & 0xFFFFu) | ((unsigned)FIN << 16)),  // dim1 hi|tile0
        16,                                           // tile_dim1=16, tile2=0
        (int)FIN,                                     // tensor_dim0_stride
        0, 0};
    i32x4 gz = {0, 0, 0, 0};
    TDM_LOAD(g0, g1, gz, gz);
    __builtin_amdgcn_s_wait_tensorcnt(0);
  }
#else
  for (int i = tid; i < 16 * FIN; i += 256) {
    int r = i / FIN, c = i - r * FIN;
    int gr = row0 + r;
    xs[r * LDSS + c] = (gr < nrows) ? X[(size_t)gr * FIN + c] : 0.f;
  }
#endif
  __syncthreads();

  const int wave = tid >> 5, lane = tid & 31;
  const int half = lane >> 4, m = lane & 15;
  constexpr int NT = FOUT / 16;  // column tiles per matrix
  for (int t = wave; t < 2 * NT; t += 8) {
    const float* W = (t < NT) ? Wl : Wr;
    float* Y = (t < NT) ? Yl : Yr;
    const int n0 = (t < NT ? t : t - NT) * 16;
    v8f acc = {};
#pragma unroll 4
    for (int k0 = 0; k0 < FIN; k0 += 4) {
      const int ka = k0 + 2 * half;
      v2f a, b;
      a.x = xs[m * LDSS + ka];
      a.y = xs[m * LDSS + ka + 1];
      b.x = W[(size_t)ka * FOUT + n0 + m];
      b.y = W[(size_t)(ka + 1) * FOUT + n0 + m];
      acc = __builtin_amdgcn_wmma_f32_16x16x4_f32(false, a, false, b,
                                                  (short)0, acc, false, false);
    }
    if (row0 + 16 <= nrows) {  // uniform fast path: no exec juggling
#pragma unroll
      for (int r = 0; r < 8; ++r)
        Y[(size_t)(row0 + half * 8 + r) * FOUT + n0 + m] = acc[r];
    } else {
#pragma unroll
      for (int r = 0; r < 8; ++r) {
        int gr = row0 + half * 8 + r;
        if (gr < nrows) Y[(size_t)gr * FOUT + n0 + m] = acc[r];
      }
    }
  }
}

// ===========================================================================
// Edge logits: e[e,h] = att[h,:] . leaky_relu(xl[src] + xr[dst]), plus
// segment-max over dst (encoded-uint atomicMax). One wave per edge; lane l
// owns NV contiguous values -> b128/b64 gathers; per-head butterfly reduce.
// ===========================================================================
template <int H>
__global__ void __launch_bounds__(256) edge_logits(
    const float* __restrict__ xl, const float* __restrict__ xr,
    const int* __restrict__ ei, const float* __restrict__ att,
    float* __restrict__ e_out, unsigned* __restrict__ m_enc, int E0, int EP) {
  int edge = blockIdx.x * 8 + (threadIdx.x >> 5);
  if (edge >= EP) return;
  int lane = threadIdx.x & 31;
  int src, dst;
  if (edge < E0) { src = ei[edge]; dst = ei[E0 + edge]; }
  else           { src = edge - E0; dst = src; }
  constexpr int CT = H * 64;
  constexpr int NV = CT / 32;   // values per lane: 8 (H=4) or 2 (H=1)
  constexpr int G = 64 / NV;    // lanes per head: 8 or 32
  const float* pl = xl + (size_t)src * CT + lane * NV;
  const float* pr = xr + (size_t)dst * CT + lane * NV;
  const float* pa = att + lane * NV;
  float partial = 0.f;
  if (NV == 8) {
#pragma unroll
    for (int i = 0; i < 2; ++i) {
      float4 a = ((const float4*)pl)[i];
      float4 b = ((const float4*)pr)[i];
      float4 w = ((const float4*)pa)[i];
      float s;
      s = a.x + b.x; partial += w.x * (s > 0.f ? s : NEG_SLOPE * s);
      s = a.y + b.y; partial += w.y * (s > 0.f ? s : NEG_SLOPE * s);
      s = a.z + b.z; partial += w.z * (s > 0.f ? s : NEG_SLOPE * s);
      s = a.w + b.w; partial += w.w * (s > 0.f ? s : NEG_SLOPE * s);
    }
  } else {
    float2 a = ((const float2*)pl)[0];
    float2 b = ((const float2*)pr)[0];
    float2 w = ((const float2*)pa)[0];
    float s;
    s = a.x + b.x; partial += w.x * (s > 0.f ? s : NEG_SLOPE * s);
    s = a.y + b.y; partial += w.y * (s > 0.f ? s : NEG_SLOPE * s);
  }
#pragma unroll
  for (int o = 1; o < G; o <<= 1) partial += __shfl_xor(partial, o, 32);
  float eh[H];
#pragma unroll
  for (int h = 0; h < H; ++h) eh[h] = __shfl(partial, h * G, 32);
  if (lane == 0) {
#pragma unroll
    for (int h = 0; h < H; ++h) {
      e_out[(size_t)edge * H + h] = eh[h];
      atomicMax(&m_enc[(size_t)dst * H + h], enc_f(eh[h]));
    }
  }
}

// a = exp(e - m[dst]); store in place; denom[dst] += a. One thread / (edge,h).
template <int H>
__global__ void __launch_bounds__(256) edge_expsum(
    const int* __restrict__ ei, float* __restrict__ e_buf,
    const unsigned* __restrict__ m_enc, float* __restrict__ denom,
    int E0, int EP) {
  int idx = blockIdx.x * blockDim.x + threadIdx.x;
  if (idx >= EP * H) return;
  int edge = idx / H;
  int h = idx - edge * H;
  int dst = (edge < E0) ? ei[E0 + edge] : (edge - E0);
  float a = expf(e_buf[idx] - dec_f(m_enc[(size_t)dst * H + h]));
  e_buf[idx] = a;
  atomicAdd(&denom[(size_t)dst * H + h], a);
}

// out[dst] += xl[src] * alpha. One wave per edge; b128 gathers, f32 atomics.
template <int H>
__global__ void __launch_bounds__(256) edge_scatter(
    const int* __restrict__ ei, const float* __restrict__ xl,
    const float* __restrict__ a_buf, const float* __restrict__ denom,
    float* __restrict__ outacc, int E0, int EP) {
  int edge = blockIdx.x * 8 + (threadIdx.x >> 5);
  if (edge >= EP) return;
  int lane = threadIdx.x & 31;
  int src, dst;
  if (edge < E0) { src = ei[edge]; dst = ei[E0 + edge]; }
  else           { src = edge - E0; dst = src; }
  constexpr int CT = H * 64;
  constexpr int NV = CT / 32;
  constexpr int G = 64 / NV;
  const int hh = lane / G;  // this lane's head
  float alpha = a_buf[(size_t)edge * H + hh] /
                (denom[(size_t)dst * H + hh] + SM_EPS);
  const float* pl = xl + (size_t)src * CT + lane * NV;
  float* po = outacc + (size_t)dst * CT + lane * NV;
  if (NV == 8) {
#pragma unroll
    for (int i = 0; i < 2; ++i) {
      float4 a = ((const float4*)pl)[i];
      atomicAdd(po + 4 * i + 0, a.x * alpha);
      atomicAdd(po + 4 * i + 1, a.y * alpha);
      atomicAdd(po + 4 * i + 2, a.z * alpha);
      atomicAdd(po + 4 * i + 3, a.w * alpha);
    }
  } else {
    float2 a = ((const float2*)pl)[0];
    atomicAdd(po + 0, a.x * alpha);
    atomicAdd(po + 1, a.y * alpha);
  }
}

// mean over heads + bias -> LayerNorm(64) -> optional ReLU. One wave per node.
template <int H, bool RELU>
__global__ void __launch_bounds__(256) finalize_ln(
    const float* __restrict__ acc, const float* __restrict__ bias,
    const float* __restrict__ gamma, const float* __restrict__ beta,
    float* __restrict__ out, int nnodes) {
  int node = blockIdx.x * 8 + (threadIdx.x >> 5);
  if (node >= nnodes) return;
  int lane = threadIdx.x & 31;
  int c0 = lane, c1 = lane + 32;
  float v0 = 0.f, v1 = 0.f;
#pragma unroll
  for (int h = 0; h < H; ++h) {
    v0 += acc[(size_t)node * H * 64 + h * 64 + c0];
    v1 += acc[(size_t)node * H * 64 + h * 64 + c1];
  }
  v0 = v0 * (1.f / H) + bias[c0];
  v1 = v1 * (1.f / H) + bias[c1];
  float mu = wave_sum(v0 + v1) * (1.f / 64.f);
  float d0 = v0 - mu, d1 = v1 - mu;
  float var = wave_sum(d0 * d0 + d1 * d1) * (1.f / 64.f);
  float inv = rsqrtf(var + LN_EPS);
  float y0 = d0 * inv * gamma[c0] + beta[c0];
  float y1 = d1 * inv * gamma[c1] + beta[c1];
  if (RELU) { y0 = fmaxf(y0, 0.f); y1 = fmaxf(y1, 0.f); }
  out[(size_t)node * 64 + c0] = y0;
  out[(size_t)node * 64 + c1] = y1;
}

// ===========================================================================
extern "C" void kernel_launch(void* const* d_in, const int* in_sizes, int n_in,
                              void* d_out, int out_size, void* d_ws,
                              size_t ws_size, hipStream_t stream) {
  (void)n_in; (void)out_size; (void)ws_size;
  const float* x    = (const float*)d_in[0];
  const int*   ei   = (const int*)d_in[1];
  const float* W1l  = (const float*)d_in[2];
  const float* W1r  = (const float*)d_in[3];
  const float* att1 = (const float*)d_in[4];
  const float* b1   = (const float*)d_in[5];
  const float* g1   = (const float*)d_in[6];
  const float* bb1  = (const float*)d_in[7];
  const float* W2l  = (const float*)d_in[8];
  const float* W2r  = (const float*)d_in[9];
  const float* att2 = (const float*)d_in[10];
  const float* b2   = (const float*)d_in[11];
  const float* g2   = (const float*)d_in[12];
  const float* bb2  = (const float*)d_in[13];

  const int N  = in_sizes[0] / 256;
  const int E  = in_sizes[1] / 2;
  const int EP = E + N;  // + self loops

  // ---- workspace carve (layer 2 reuses layer 1 buffers) ----
  char* p = (char*)d_ws;
  auto carve = [&](size_t bytes) -> char* {
    char* r = p;
    p += (bytes + 255) & ~(size_t)255;
    return r;
  };
  float*    xl1  = (float*)carve((size_t)N * 256 * 4);  // also xl2
  float*    xr1  = (float*)carve((size_t)N * 256 * 4);  // also xr2
  float*    acc  = (float*)carve((size_t)N * 256 * 4);  // segment-sum accum
  float*    ebuf = (float*)carve((size_t)EP * 4 * 4);   // logits -> exp(e-m)
  unsigned* menc = (unsigned*)carve((size_t)N * 4 * 4); // encoded segment max
  float*    den  = (float*)carve((size_t)N * 4 * 4);    // softmax denominator
  float*    hbuf = (float*)carve((size_t)N * 64 * 4);   // layer-1 output

  const int rowBlocks = (N + 15) / 16;
  const int edgeBlocks = (EP + 7) / 8;
  const int nodeBlocks = (N + 7) / 8;

  // ---------------- Layer 1 (H=4, C=64) ----------------
  hipMemsetAsync(menc, 0, (size_t)N * 4 * sizeof(unsigned), stream);
  hipMemsetAsync(den,  0, (size_t)N * 4 * sizeof(float), stream);
  hipMemsetAsync(acc,  0, (size_t)N * 256 * sizeof(float), stream);
  gemm_pair<256, 256><<<rowBlocks, 256, 0, stream>>>(x, W1l, W1r, xl1, xr1, N);
  edge_logits<4><<<edgeBlocks, 256, 0, stream>>>(xl1, xr1, ei, att1, ebuf,
                                                 menc, E, EP);
  edge_expsum<4><<<(EP * 4 + 255) / 256, 256, 0, stream>>>(ei, ebuf, menc,
                                                           den, E, EP);
  edge_scatter<4><<<edgeBlocks, 256, 0, stream>>>(ei, xl1, ebuf, den, acc,
                                                  E, EP);
  finalize_ln<4, true><<<nodeBlocks, 256, 0, stream>>>(acc, b1, g1, bb1,
                                                       hbuf, N);

  // ---------------- Layer 2 (H=1, C=64) ----------------
  hipMemsetAsync(menc, 0, (size_t)N * sizeof(unsigned), stream);
  hipMemsetAsync(den,  0, (size_t)N * sizeof(float), stream);
  hipMemsetAsync(acc,  0, (size_t)N * 64 * sizeof(float), stream);
  gemm_pair<64, 64><<<rowBlocks, 256, 0, stream>>>(hbuf, W2l, W2r, xl1, xr1, N);
  edge_logits<1><<<edgeBlocks, 256, 0, stream>>>(xl1, xr1, ei, att2, ebuf,
                                                 menc, E, EP);
  edge_expsum<1><<<(EP + 255) / 256, 256, 0, stream>>>(ei, ebuf, menc, den,
                                                       E, EP);
  edge_scatter<1><<<edgeBlocks, 256, 0, stream>>>(ei, xl1, ebuf, den, acc,
                                                  E, EP);
  finalize_ln<1, false><<<nodeBlocks, 256, 0, stream>>>(acc, b2, g2, bb2,
                                                        (float*)d_out, N);
}